// DynamicMaskHead_13812614824606
// MI455X (gfx1250) — compile-verified
//
#include <hip/hip_runtime.h>
#include <hip/hip_bf16.h>

typedef __attribute__((ext_vector_type(2))) float v2f;
typedef __attribute__((ext_vector_type(8))) float v8f;

#define HH 96
#define WW 160
#define HW (HH * WW)
#define NINST 200
#define NPAR 169
#define OH 192
#define OW 320

__device__ __forceinline__ v8f wmma_f32_4(v2f a, v2f b, v8f c) {
  // D = A(16x4 f32) * B(4x16 f32) + C(16x16 f32)
  return __builtin_amdgcn_wmma_f32_16x16x4_f32(false, a, false, b, (short)0, c,
                                               false, false);
}

__global__ __launch_bounds__(256) void dynmask_fused_kernel(
    const float* __restrict__ mask_feats,   // [2,8,96,160]
    const float* __restrict__ params0,      // [200,169]
    const float* __restrict__ params1,      // [200,169]
    const float* __restrict__ locs,         // [200,2]
    const float* __restrict__ offs,         // [200,2]
    const int* __restrict__ im_inds,        // [200]
    const int* __restrict__ fpn_levels,     // [200]
    const int* __restrict__ stride_p,       // [1]
    float* __restrict__ out)                // [2,200,192,320]
{
  __shared__ float lds[9 * WW];

  const int inst = blockIdx.x;
  const int band = blockIdx.y;
  const int head = blockIdx.z;

  const int lane = threadIdx.x & 31;
  const int wave = threadIdx.x >> 5;
  const int nwaves = blockDim.x >> 5;
  const int Mrow = lane & 15;   // M (out-channel) index of this lane for A
  const int kh   = lane >> 4;   // K half-select (0: K=0,1 ; 1: K=2,3)

  const float* P = (head ? params1 : params0) + inst * NPAR;
  const int im  = im_inds[inst];
  const int lvl = fpn_levels[inst];
  const float inv_soi = 1.0f / (float)(64 << lvl);  // SOI = 64*2^lvl

  float lx = locs[inst * 2 + 0];
  float ly = locs[inst * 2 + 1];
  if (head) {
    lx += offs[inst * 2 + 0] * 128.0f;
    ly += offs[inst * 2 + 1] * 128.0f;
  }
  const int   stride = stride_p[0];
  const float center = (float)(stride >> 1);

  // ---- load per-instance weights into WMMA A fragments (zero-padded) ----
  v2f A0[3];  // layer0: W0 [8x10] padded to 16x12 (3 K-chunks of 4)
#pragma unroll
  for (int kc = 0; kc < 3; kc++) {
#pragma unroll
    for (int v = 0; v < 2; v++) {
      const int K = 4 * kc + 2 * kh + v;
      float w = 0.0f;
      if (Mrow < 8 && K < 10) w = P[Mrow * 10 + K];
      A0[kc][v] = w;
    }
  }
  v2f A1[2];  // layer1: W1 [8x8] padded to 16x8 (2 K-chunks of 4)
#pragma unroll
  for (int kc = 0; kc < 2; kc++) {
#pragma unroll
    for (int v = 0; v < 2; v++) {
      const int K = 4 * kc + 2 * kh + v;
      A1[kc][v] = (Mrow < 8) ? P[80 + Mrow * 8 + K] : 0.0f;
    }
  }
  float b0v[8], b1v[8], w2v[8];
#pragma unroll
  for (int v = 0; v < 8; v++) {
    w2v[v] = P[144 + v];
    b0v[v] = P[152 + v];
    b1v[v] = P[160 + v];
  }
  const float b2s = P[168];

  // ---- phase 1: per-band logits via WMMA into LDS ----
  const int base   = 8 * band - 1;                 // lds row 0 == input row `base`
  const int ystart = (band == 0) ? 0 : base;
  const int nrows  = (band == 0) ? 8 : 9;
  const int ldsoff = (ystart - base) * WW;         // 160 for band 0, else 0
  const int ntiles = (nrows * WW) >> 4;            // 80 or 90 full tiles
  const float* fbase = mask_feats + (size_t)im * 8 * HW;

  for (int t = wave; t < ntiles; t += nwaves) {
    const int p = t * 16 + (lane & 15);            // pixel id within band
    const int y = ystart + p / WW;
    const int x = p % WW;
    const float* f = fbase + y * WW + x;

    const float relx = (lx - (float)(x * stride) - center) * inv_soi;
    const float rely = (ly - (float)(y * stride) - center) * inv_soi;

    // B fragments: channels K = [relx, rely, f0..f7, 0, 0]
    v2f B0, B1, B2;
    if (kh == 0) {            // K = {0,1},{4,5},{8,9}
      B0[0] = relx;       B0[1] = rely;
      B1[0] = f[2 * HW];  B1[1] = f[3 * HW];
      B2[0] = f[6 * HW];  B2[1] = f[7 * HW];
    } else {                  // K = {2,3},{6,7},{10,11}
      B0[0] = f[0 * HW];  B0[1] = f[1 * HW];
      B1[0] = f[4 * HW];  B1[1] = f[5 * HW];
      B2[0] = 0.0f;       B2[1] = 0.0f;
    }

    // layer 0: D = W0 * x + b0, relu
    v8f c;
#pragma unroll
    for (int v = 0; v < 8; v++) c[v] = kh ? 0.0f : b0v[v];  // M = v + 8*kh
    c = wmma_f32_4(A0[0], B0, c);
    c = wmma_f32_4(A0[1], B1, c);
    c = wmma_f32_4(A0[2], B2, c);
#pragma unroll
    for (int v = 0; v < 8; v++) c[v] = fmaxf(c[v], 0.0f);

    // D layout -> layer1 B layout: gather M=0..7 of this pixel from lane&15
    float tj[8];
#pragma unroll
    for (int j = 0; j < 8; j++) tj[j] = __shfl(c[j], lane & 15, 32);
    v2f C0, C1;
    C0[0] = kh ? tj[2] : tj[0];  C0[1] = kh ? tj[3] : tj[1];
    C1[0] = kh ? tj[6] : tj[4];  C1[1] = kh ? tj[7] : tj[5];

    // layer 1: D = W1 * relu0 + b1
    v8f d;
#pragma unroll
    for (int v = 0; v < 8; v++) d[v] = kh ? 0.0f : b1v[v];
    d = wmma_f32_4(A1[0], C0, d);
    d = wmma_f32_4(A1[1], C1, d);

    // layer 2: scalar dot over the 8 M-values this lane holds (valid for kh==0)
    float s = b2s;
#pragma unroll
    for (int v = 0; v < 8; v++) s += w2v[v] * fmaxf(d[v], 0.0f);
    if (kh == 0) lds[ldsoff + p] = s;
  }
  __syncthreads();

  // ---- phase 2: aligned_bilinear x2 + sigmoid, stream 16x320 output tile ----
  const int r0 = 16 * band;
  const size_t obase = ((size_t)head * NINST + inst) * (size_t)(OH * OW);
  for (int idx = threadIdx.x; idx < 16 * OW; idx += blockDim.x) {
    const int r   = r0 + idx / OW;
    const int col = idx % OW;
    int R = r - 1;   if (R < 0) R = 0;
    int C = col - 1; if (C < 0) C = 0;
    const int y0 = R >> 1;
    const int x0 = C >> 1;
    const float wy = (R & 1) ? 0.5f : 0.0f;
    const float wx = (C & 1) ? 0.5f : 0.0f;
    int y1 = y0, x1 = x0;
    if (R & 1) { y1 = y0 + 1; if (y1 > HH - 1) y1 = HH - 1; }
    if (C & 1) { x1 = x0 + 1; if (x1 > WW - 1) x1 = WW - 1; }
    const float v00 = lds[(y0 - base) * WW + x0];
    const float v01 = lds[(y0 - base) * WW + x1];
    const float v10 = lds[(y1 - base) * WW + x0];
    const float v11 = lds[(y1 - base) * WW + x1];
    const float top = v00 + wx * (v01 - v00);
    const float bot = v10 + wx * (v11 - v10);
    const float val = top + wy * (bot - top);
    out[obase + (size_t)r * OW + col] = 1.0f / (1.0f + __expf(-val));
  }
}

extern "C" void kernel_launch(void* const* d_in, const int* in_sizes, int n_in,
                              void* d_out, int out_size, void* d_ws, size_t ws_size,
                              hipStream_t stream) {
  const float* mask_feats = (const float*)d_in[0];
  const float* params0    = (const float*)d_in[1];
  const float* params1    = (const float*)d_in[2];
  const float* locs       = (const float*)d_in[3];
  const float* offs       = (const float*)d_in[4];
  const int*   im_inds    = (const int*)d_in[5];
  const int*   lvls       = (const int*)d_in[6];
  const int*   stridep    = (const int*)d_in[7];

  dim3 grid(NINST, HH / 8, 2);   // instance x row-band x head
  dim3 block(256, 1, 1);
  hipLaunchKernelGGL(dynmask_fused_kernel, grid, block, 0, stream,
                     mask_feats, params0, params1, locs, offs,
                     im_inds, lvls, stridep, (float*)d_out);
}